// ProductLayer_8065948581965
// MI455X (gfx1250) — compile-verified
//
#include <hip/hip_runtime.h>

typedef __attribute__((ext_vector_type(2))) float v2f;
typedef __attribute__((ext_vector_type(8))) float v8f;

#define B_N 4096
#define F_N 32
#define D_N 64
#define U_N 256
#define K_LZ (F_N * D_N)        // 2048
#define OUT_STRIDE (3 * U_N)    // 768

__device__ __forceinline__ v8f wmma4(v2f a, v2f b, v8f c) {
  // D = A(16x4,f32) * B(4x16,f32) + C(16x16,f32)
  return __builtin_amdgcn_wmma_f32_16x16x4_f32(
      /*neg_a=*/false, a, /*neg_b=*/false, b,
      /*c_mod=*/(short)0, c, /*reuse_a=*/false, /*reuse_b=*/false);
}

// ---------------------------------------------------------------------------
// fs[b,d] = sum_f embeds[b,f,d]   (1 MB workspace)
// ---------------------------------------------------------------------------
__global__ void __launch_bounds__(256) fs_kernel(const float* __restrict__ embeds,
                                                 float* __restrict__ fs) {
  int idx = blockIdx.x * 256 + threadIdx.x;   // 0 .. B*D-1
  if (idx >= B_N * D_N) return;
  int b = idx >> 6, d = idx & 63;
  const float* e = embeds + (size_t)b * K_LZ + d;
  float s = 0.f;
#pragma unroll
  for (int f = 0; f < F_N; ++f) s += e[f * D_N];
  fs[idx] = s;
}

// ---------------------------------------------------------------------------
// lz: [4096 x 2048] @ [2048 x 256] -> out[:, 0:256]
// Block: M=32, N=256. Wave: 2 M-tiles x 2 N-tiles = 4 independent WMMA chains;
// each A/B operand fetch feeds 2 WMMAs. A chunk staged in LDS.
// ---------------------------------------------------------------------------
#define LZ_KC 128
#define LZ_AP 132   // pad: lane-halves (K, K+2) hit disjoint bank sets
__global__ void __launch_bounds__(256) lz_kernel(const float* __restrict__ embeds,
                                                 const float* __restrict__ linear_w,
                                                 float* __restrict__ out) {
  __shared__ float ldsA[32 * LZ_AP];
  const int tid = threadIdx.x;
  const int lane = tid & 31;
  const int wave = tid >> 5;
  const int bt = blockIdx.x;                  // 32-row M tile
  const int m = lane & 15;
  const int koff = (lane < 16) ? 0 : 2;
  const int n0 = wave * 16 + m;               // N-tile #wave       (u 0..127)
  const int n1 = (wave + 8) * 16 + m;         // N-tile #wave+8     (u 128..255)
  v8f acc00 = {}, acc01 = {}, acc10 = {}, acc11 = {};

  for (int kb = 0; kb < K_LZ; kb += LZ_KC) {
    __syncthreads();
    for (int i = tid; i < 32 * LZ_KC; i += 256) {
      int r = i >> 7, c = i & (LZ_KC - 1);
      const float* src = embeds + (size_t)(bt * 32 + r) * K_LZ + kb + c;
      ldsA[r * LZ_AP + c] = *src;
      if (kb + LZ_KC < K_LZ) __builtin_prefetch(src + LZ_KC, 0, 0);
    }
    __syncthreads();
#pragma unroll 8
    for (int kk = 0; kk < LZ_KC; kk += 4) {
      v2f a0, a1;
      a0.x = ldsA[m * LZ_AP + kk + koff];
      a0.y = ldsA[m * LZ_AP + kk + koff + 1];
      a1.x = ldsA[(m + 16) * LZ_AP + kk + koff];
      a1.y = ldsA[(m + 16) * LZ_AP + kk + koff + 1];
      const float* bp = linear_w + (size_t)(kb + kk + koff) * U_N;
      v2f b0; b0.x = bp[n0]; b0.y = bp[U_N + n0];
      v2f b1; b1.x = bp[n1]; b1.y = bp[U_N + n1];
      acc00 = wmma4(a0, b0, acc00);
      acc01 = wmma4(a0, b1, acc01);
      acc10 = wmma4(a1, b0, acc10);
      acc11 = wmma4(a1, b1, acc11);
    }
  }
  const int rb0 = bt * 32 + ((lane < 16) ? 0 : 8);
  const int rb1 = rb0 + 16;
#pragma unroll
  for (int r = 0; r < 8; ++r) {
    out[(size_t)(rb0 + r) * OUT_STRIDE + n0] = acc00[r];
    out[(size_t)(rb0 + r) * OUT_STRIDE + n1] = acc01[r];
    out[(size_t)(rb1 + r) * OUT_STRIDE + n0] = acc10[r];
    out[(size_t)(rb1 + r) * OUT_STRIDE + n1] = acc11[r];
  }
}

// ---------------------------------------------------------------------------
// lp_inner: per b, delta[d,u] = A(d,f) @ B(f,u), A = embeds[b]^T (LDS),
// B = inner_w^T; lp_inner[b,u] = sum_d delta^2 -> out[:, 256:512]
// Wave: 2 u-tiles x 4 d-tiles (4 chains share each B operand), K=F=32.
// ---------------------------------------------------------------------------
#define IN_EP 72   // f-stride pad: halves (f, f+2) -> bank sets offset by 16
__global__ void __launch_bounds__(256) inner_kernel(const float* __restrict__ embeds,
                                                    const float* __restrict__ inner_w,
                                                    float* __restrict__ out) {
  __shared__ float ldsE[F_N * IN_EP];
  const int b = blockIdx.x;
  const int tid = threadIdx.x;
  const int lane = tid & 31;
  const int wave = tid >> 5;
  for (int i = tid; i < F_N * D_N; i += 256) {
    int f = i >> 6, d = i & 63;
    ldsE[f * IN_EP + d] = embeds[(size_t)b * K_LZ + i];
  }
  __syncthreads();
  const int koff = (lane < 16) ? 0 : 2;
  const int ln = lane & 15;
  const v8f vzero = {};
#pragma unroll
  for (int nt = 0; nt < 2; ++nt) {
    const int u = wave * 32 + nt * 16 + ln;
    v8f acc[4];
#pragma unroll
    for (int mt = 0; mt < 4; ++mt) acc[mt] = vzero;
#pragma unroll
    for (int fk = 0; fk < F_N; fk += 4) {
      v2f bv;
      bv.x = inner_w[u * F_N + fk + koff];
      bv.y = inner_w[u * F_N + fk + koff + 1];
#pragma unroll
      for (int mt = 0; mt < 4; ++mt) {     // 4 independent d-tiles share bv
        int d = mt * 16 + ln;
        v2f av;
        av.x = ldsE[(fk + koff) * IN_EP + d];
        av.y = ldsE[(fk + koff + 1) * IN_EP + d];
        acc[mt] = wmma4(av, bv, acc[mt]);
      }
    }
    float p = 0.f;
#pragma unroll
    for (int mt = 0; mt < 4; ++mt)
#pragma unroll
      for (int r = 0; r < 8; ++r) p += acc[mt][r] * acc[mt][r];
    p += __shfl_xor(p, 16, 32);            // fold rows 8..15 half into 0..7 half
    if (lane < 16) out[(size_t)b * OUT_STRIDE + U_N + u] = p;
  }
}

// ---------------------------------------------------------------------------
// lp_outer as GEMM: O[b, i*64+j] = fs[b,i]*fs[b,j] built on the fly from LDS,
// Wt[k,u] = outer_w[u,k] -> out[:, 512:768].
// Block: M=32, N=256. Wave: 2 M-tiles x 2 N-tiles (4 chains; B shared over M).
// ---------------------------------------------------------------------------
#define FS_P 68
__global__ void __launch_bounds__(256) outer_kernel(const float* __restrict__ fs,
                                                    const float* __restrict__ outer_w,
                                                    float* __restrict__ out) {
  __shared__ float ldsF[32 * FS_P];
  const int bt = blockIdx.x;                  // 32-row M tile
  const int tid = threadIdx.x;
  const int lane = tid & 31;
  const int wave = tid >> 5;
  for (int i = tid; i < 32 * D_N; i += 256) {
    int r = i >> 6, c = i & 63;
    ldsF[r * FS_P + c] = fs[(size_t)(bt * 32 + r) * D_N + c];
  }
  __syncthreads();
  const int koff = (lane < 16) ? 0 : 2;
  const int m = lane & 15;
  const int u0 = wave * 16 + m;
  const int u1 = (wave + 8) * 16 + m;
  const float* w0 = outer_w + (size_t)u0 * (D_N * D_N);
  const float* w1 = outer_w + (size_t)u1 * (D_N * D_N);
  v8f acc00 = {}, acc01 = {}, acc10 = {}, acc11 = {};
  for (int i = 0; i < D_N; ++i) {
    const float fi0 = ldsF[m * FS_P + i];        // broadcast reads
    const float fi1 = ldsF[(m + 16) * FS_P + i];
    __builtin_prefetch(w0 + (i + 4) * D_N, 0, 0);   // speculative stream-ahead
    __builtin_prefetch(w1 + (i + 4) * D_N, 0, 0);
#pragma unroll
    for (int j = 0; j < D_N; j += 4) {
      const int k = i * D_N + j + koff;        // even -> 8B-aligned v2f loads
      v2f a0, a1;
      a0.x = fi0 * ldsF[m * FS_P + j + koff];
      a0.y = fi0 * ldsF[m * FS_P + j + koff + 1];
      a1.x = fi1 * ldsF[(m + 16) * FS_P + j + koff];
      a1.y = fi1 * ldsF[(m + 16) * FS_P + j + koff + 1];
      v2f b0 = *(const v2f*)(w0 + k);
      v2f b1 = *(const v2f*)(w1 + k);
      acc00 = wmma4(a0, b0, acc00);
      acc01 = wmma4(a0, b1, acc01);
      acc10 = wmma4(a1, b0, acc10);
      acc11 = wmma4(a1, b1, acc11);
    }
  }
  const int rb0 = bt * 32 + ((lane < 16) ? 0 : 8);
  const int rb1 = rb0 + 16;
#pragma unroll
  for (int r = 0; r < 8; ++r) {
    out[(size_t)(rb0 + r) * OUT_STRIDE + 2 * U_N + u0] = acc00[r];
    out[(size_t)(rb0 + r) * OUT_STRIDE + 2 * U_N + u1] = acc01[r];
    out[(size_t)(rb1 + r) * OUT_STRIDE + 2 * U_N + u0] = acc10[r];
    out[(size_t)(rb1 + r) * OUT_STRIDE + 2 * U_N + u1] = acc11[r];
  }
}

extern "C" void kernel_launch(void* const* d_in, const int* in_sizes, int n_in,
                              void* d_out, int out_size, void* d_ws, size_t ws_size,
                              hipStream_t stream) {
  const float* embeds   = (const float*)d_in[0];   // [B, F, D]
  const float* linear_w = (const float*)d_in[1];   // [F*D, U]
  const float* inner_w  = (const float*)d_in[2];   // [U, F]
  const float* outer_w  = (const float*)d_in[3];   // [U, D, D]
  float* out = (float*)d_out;                      // [B, 3U]
  float* fs  = (float*)d_ws;                       // [B, D] scratch (1 MB)

  fs_kernel<<<(B_N * D_N + 255) / 256, 256, 0, stream>>>(embeds, fs);
  lz_kernel<<<B_N / 32, 256, 0, stream>>>(embeds, linear_w, out);
  inner_kernel<<<B_N, 256, 0, stream>>>(embeds, inner_w, out);
  outer_kernel<<<B_N / 32, 256, 0, stream>>>(fs, outer_w, out);
}